// ConvLayer_6279242186805
// MI455X (gfx1250) — compile-verified
//
#include <hip/hip_runtime.h>
#include <hip/hip_bf16.h>

typedef __attribute__((ext_vector_type(2))) float v2f;
typedef __attribute__((ext_vector_type(8))) float v8f;

#define LRELU_SLOPE 0.01f
#define BN_EPS 1e-5f

// ---------------------------------------------------------------------------
// Zero output accumulator + BN stats scratch (harness poisons with 0xAA once,
// and we atomically accumulate, so we must zero on every call).
// ---------------------------------------------------------------------------
__global__ void zero_kernel(float* __restrict__ out, float* __restrict__ stats,
                            long long n) {
    long long i = (long long)blockIdx.x * blockDim.x + threadIdx.x;
    const long long stride = (long long)gridDim.x * blockDim.x;
    if (i < 64) stats[i] = 0.0f;
    for (; i < n; i += stride) out[i] = 0.0f;
}

// ---------------------------------------------------------------------------
// Sparse submanifold conv via FP32 WMMA.
// grid = (blocksX, K); block = 256 (8 waves). Each wave processes
// tilesPerWave tiles of 16 pairs for kernel-offset k = blockIdx.y.
// Per tile: gather A (16x32), 16x v_wmma_f32_16x16x4_f32, atomic scatter-add.
// ---------------------------------------------------------------------------
__global__ void __launch_bounds__(256)
subm_conv_wmma(const float* __restrict__ features,
               const float* __restrict__ weight,
               const int*   __restrict__ pairs_in,
               const int*   __restrict__ pairs_out,
               float*       __restrict__ out,
               int P, int tilesPerK, int tilesPerWave) {
    const int k    = blockIdx.y;
    const int lane = threadIdx.x & 31;
    const int wid  = threadIdx.x >> 5;
    const int half = lane >> 4;   // 0: lanes 0-15, 1: lanes 16-31
    const int mIdx = lane & 15;

    // ---- Preload B = W[k] (Cin=32 x Cout=32, row-major, cout stride 1).
    // Per K-chunk kc: B tile is 4x16. VGPR0 holds rows K0 (lanes0-15) / K2
    // (lanes16-31), VGPR1 holds K1 / K3 (mirrors A-matrix layout).
    const float* wk = weight + (size_t)k * 32 * 32;
    v2f bLo[8], bHi[8];
#pragma unroll
    for (int kc = 0; kc < 8; ++kc) {
        const int r0 = kc * 4 + 2 * half;
        bLo[kc].x = wk[(r0    ) * 32 + mIdx];
        bLo[kc].y = wk[(r0 + 1) * 32 + mIdx];
        bHi[kc].x = wk[(r0    ) * 32 + 16 + mIdx];
        bHi[kc].y = wk[(r0 + 1) * 32 + 16 + mIdx];
    }

    const int waveGlobal = blockIdx.x * 8 + wid;
    int t0 = waveGlobal * tilesPerWave;
    int t1 = t0 + tilesPerWave;
    if (t1 > tilesPerK) t1 = tilesPerK;
    const long long kp = (long long)k * P;

    for (int t = t0; t < t1; ++t) {
        const int base   = t * 16;
        const int mrow   = base + mIdx;
        const bool mvalid = (mrow < P);

        int inRow = 0;
        if (mvalid) inRow = pairs_in[kp + mrow];
        const float* fr = features + (size_t)inRow * 32;

        v8f accLo = {};
        v8f accHi = {};
#pragma unroll
        for (int kc = 0; kc < 8; ++kc) {
            // A 16x4 tile: lane holds feat[row][c0], feat[row][c0+1]
            // (VGPR0 = K0/K2 across halves, VGPR1 = K1/K3) -> one b64 load.
            const int c0 = kc * 4 + 2 * half;
            v2f a;
            if (mvalid) a = *reinterpret_cast<const v2f*>(fr + c0);
            else { a.x = 0.0f; a.y = 0.0f; }
            accLo = __builtin_amdgcn_wmma_f32_16x16x4_f32(
                false, a, false, bLo[kc], (short)0, accLo, false, false);
            accHi = __builtin_amdgcn_wmma_f32_16x16x4_f32(
                false, a, false, bHi[kc], (short)0, accHi, false, false);
        }

        // ---- Scatter-add. D layout: VGPR v -> M = v + 8*half, N = mIdx.
#pragma unroll
        for (int v = 0; v < 8; ++v) {
            const int r = base + v + 8 * half;
            if (r < P) {
                const int orow = pairs_out[kp + r];
                float* op = out + (size_t)orow * 32 + mIdx;
                atomicAdd(op,      accLo[v]);
                atomicAdd(op + 16, accHi[v]);
            }
        }
    }
}

// ---------------------------------------------------------------------------
// Per-channel sum / sum-of-squares of LeakyReLU(conv). blockDim=256 so the
// grid stride is a multiple of 32 -> each thread owns one fixed channel.
// ---------------------------------------------------------------------------
__global__ void __launch_bounds__(256)
stats_kernel(const float* __restrict__ conv, float* __restrict__ stats,
             long long n) {
    __shared__ float sS[256];
    __shared__ float sQ[256];
    const int tid = threadIdx.x;
    long long i = (long long)blockIdx.x * blockDim.x + tid;
    const long long stride = (long long)gridDim.x * blockDim.x;
    float s = 0.0f, q = 0.0f;
    for (; i < n; i += stride) {
        const float x = conv[i];
        const float a = (x >= 0.0f) ? x : LRELU_SLOPE * x;
        s += a;
        q += a * a;
    }
    sS[tid] = s;
    sQ[tid] = q;
    __syncthreads();
    for (int off = 128; off >= 32; off >>= 1) {
        if (tid < off) { sS[tid] += sS[tid + off]; sQ[tid] += sQ[tid + off]; }
        __syncthreads();
    }
    if (tid < 32) {
        atomicAdd(&stats[tid],      sS[tid]);
        atomicAdd(&stats[32 + tid], sQ[tid]);
    }
}

// ---------------------------------------------------------------------------
// Fused LeakyReLU + BatchNorm (training stats), in place on d_out.
// ---------------------------------------------------------------------------
__global__ void __launch_bounds__(256)
bn_kernel(float* __restrict__ out, const float* __restrict__ stats,
          const float* __restrict__ gamma, const float* __restrict__ beta,
          long long n, float invN) {
    long long i = (long long)blockIdx.x * blockDim.x + threadIdx.x;
    const long long stride = (long long)gridDim.x * blockDim.x;
    for (; i < n; i += stride) {
        const int c = (int)(i & 31);
        const float x = out[i];
        const float a = (x >= 0.0f) ? x : LRELU_SLOPE * x;
        const float mean = stats[c] * invN;
        const float var  = stats[32 + c] * invN - mean * mean;
        out[i] = (a - mean) * rsqrtf(var + BN_EPS) * gamma[c] + beta[c];
    }
}

// ---------------------------------------------------------------------------
extern "C" void kernel_launch(void* const* d_in, const int* in_sizes, int n_in,
                              void* d_out, int out_size, void* d_ws, size_t ws_size,
                              hipStream_t stream) {
    const float* features = (const float*)d_in[0];
    const float* weight   = (const float*)d_in[1];
    const float* gamma    = (const float*)d_in[2];
    const float* beta     = (const float*)d_in[3];
    const int*   pairs_in  = (const int*)d_in[4];
    const int*   pairs_out = (const int*)d_in[5];
    float* out   = (float*)d_out;
    float* stats = (float*)d_ws;   // 64 floats: [0..31]=sum, [32..63]=sumsq

    const int Cin = 32, Cout = 32;
    const long long N = (long long)in_sizes[0] / Cin;
    const int K = in_sizes[1] / (Cin * Cout);
    const int P = in_sizes[4] / K;
    const long long n = N * Cout;   // == out_size

    // 1) zero accumulator + stats
    {
        long long b = (n + 255) / 256;
        if (b > 16384) b = 16384;
        zero_kernel<<<dim3((unsigned)b), 256, 0, stream>>>(out, stats, n);
    }

    // 2) sparse conv with FP32 WMMA + atomic scatter
    {
        const int tilesPerK = (P + 15) / 16;
        const int TPW = 8;                                  // tiles per wave
        const int wavesPerK = (tilesPerK + TPW - 1) / TPW;
        const int blocksX = (wavesPerK + 7) / 8;            // 8 waves / block
        subm_conv_wmma<<<dim3(blocksX, K), 256, 0, stream>>>(
            features, weight, pairs_in, pairs_out, out, P, tilesPerK, TPW);
    }

    // 3) BN statistics over LeakyReLU(conv)
    stats_kernel<<<dim3(1024), 256, 0, stream>>>(out, stats, n);

    // 4) fused LeakyReLU + BN normalize, in place
    {
        long long b = (n + 255) / 256;
        if (b > 65535) b = 65535;
        bn_kernel<<<dim3((unsigned)b), 256, 0, stream>>>(
            out, stats, gamma, beta, n, 1.0f / (float)N);
    }
}